// GraphAutoEncoder_2473901162775
// MI455X (gfx1250) — compile-verified
//
#include <hip/hip_runtime.h>
#include <math.h>

// Problem constants (match reference)
#define BATCH   64
#define NPT     128
#define IND     5
#define LATD    3
#define HIDD    64
#define ALPHA_C 0.1f
#define NEG_SL  0.2f

typedef __attribute__((ext_vector_type(2))) float v2f;
typedef __attribute__((ext_vector_type(8))) float v8f;

// ---------------------------------------------------------------------------
// Encoder layer 1: h1 = relu(batch @ W1 + b1), K=5 (scalar; too thin for WMMA)
// Also copies batch[:,:,:4] and batch[:,:,4:5] to output slices.
// grid = B, block = 128 (one thread per node row)
// ---------------------------------------------------------------------------
__global__ void enc1_kernel(const float* __restrict__ batch,
                            const float* __restrict__ w1, const float* __restrict__ b1,
                            float* __restrict__ h1,
                            float* __restrict__ o_feat4, float* __restrict__ o_feat1) {
  __shared__ float Ws[IND * HIDD];
  __shared__ float Bs[HIDD];
  const int b = blockIdx.x;
  const int t = threadIdx.x;
  for (int i = t; i < IND * HIDD; i += 128) Ws[i] = w1[i];
  if (t < HIDD) Bs[t] = b1[t];
  __syncthreads();
  float x[IND];
#pragma unroll
  for (int k = 0; k < IND; ++k) x[k] = batch[(b * NPT + t) * IND + k];
#pragma unroll
  for (int k = 0; k < 4; ++k) o_feat4[(b * NPT + t) * 4 + k] = x[k];
  o_feat1[b * NPT + t] = x[4];
  for (int c = 0; c < HIDD; ++c) {
    float acc = Bs[c];
#pragma unroll
    for (int k = 0; k < IND; ++k) acc = fmaf(x[k], Ws[k * HIDD + c], acc);
    h1[(b * NPT + t) * HIDD + c] = fmaxf(acc, 0.0f);
  }
}

// ---------------------------------------------------------------------------
// WMMA GEMM: out[b] = epilogue( A[b](128x64) @ W(64x64) + bias [+ alpha*skip] )
// fp32 WMMA 16x16x4. grid = B, block = 256 (8 waves).
// Per wave: N-tile fixed (nt = w & 3); 4 M-tiles. B fragments + bias preloaded
// once; per M-tile all 16 A fragments are preloaded so the 16 WMMAs issue
// back-to-back with a single ds-wait in front.
// ---------------------------------------------------------------------------
template <bool RELU, bool SKIP>
__global__ __launch_bounds__(256) void gemm_k64_wmma(
    const float* __restrict__ A, const float* __restrict__ W,
    const float* __restrict__ bias, const float* __restrict__ skip,
    float alpha, float* __restrict__ out) {
  __shared__ float As[NPT * 65];
  __shared__ float Wsh[HIDD * 65];
  __shared__ float Bs[HIDD];
  const int b = blockIdx.x;
  const int t = threadIdx.x;
  const float* Ab = A + (size_t)b * NPT * HIDD;
  for (int i = t; i < NPT * HIDD; i += 256) As[(i >> 6) * 65 + (i & 63)] = Ab[i];
  for (int i = t; i < HIDD * HIDD; i += 256) Wsh[(i >> 6) * 65 + (i & 63)] = W[i];
  if (t < HIDD) Bs[t] = bias[t];
  __syncthreads();

  const int w = t >> 5, lane = t & 31;
  const int rlo = lane & 15, half = lane >> 4;
  const int nt = w & 3;                 // N-tile fixed per wave
  const int col = nt * 16 + rlo;

  // Preload all 16 B fragments (loop-invariant across M-tiles) and bias.
  v2f bfr[16];
#pragma unroll
  for (int kt = 0; kt < 16; ++kt) {
    const int kb = kt * 4 + half * 2;
    bfr[kt][0] = Wsh[(kb + 0) * 65 + col];
    bfr[kt][1] = Wsh[(kb + 1) * 65 + col];
  }
  const float bv = Bs[col];

#pragma unroll
  for (int mi = 0; mi < 4; ++mi) {
    const int mt = (w >> 2) + 2 * mi;   // 4 M-tiles per wave
    const int arow = mt * 16 + rlo;
    // Batch all A-fragment LDS loads, then a single WMMA chain.
    v2f afr[16];
#pragma unroll
    for (int kt = 0; kt < 16; ++kt) {
      const int kb = kt * 4 + half * 2;
      afr[kt][0] = As[arow * 65 + kb + 0];
      afr[kt][1] = As[arow * 65 + kb + 1];
    }
    v8f c;
#pragma unroll
    for (int r = 0; r < 8; ++r) c[r] = bv;   // bias fused into accumulator
#pragma unroll
    for (int kt = 0; kt < 16; ++kt)
      c = __builtin_amdgcn_wmma_f32_16x16x4_f32(false, afr[kt], false, bfr[kt],
                                                (short)0, c, false, false);
#pragma unroll
    for (int r = 0; r < 8; ++r) {
      const int row = mt * 16 + half * 8 + r;
      float v = c[r];
      if (SKIP) v = fmaf(alpha, skip[((size_t)b * NPT + row) * HIDD + col], v);
      if (RELU) v = fmaxf(v, 0.0f);
      out[((size_t)b * NPT + row) * HIDD + col] = v;
    }
  }
}

// ---------------------------------------------------------------------------
// Latent + skip: latent = h2 @ W3 + b3 (64->3); skip = latent @ Wsk + bsk (3->64)
// grid = B, block = 128
// ---------------------------------------------------------------------------
__global__ void latent_skip_kernel(const float* __restrict__ h2,
                                   const float* __restrict__ w3, const float* __restrict__ b3,
                                   const float* __restrict__ skw, const float* __restrict__ skb,
                                   float* __restrict__ latent, float* __restrict__ skip) {
  __shared__ float W3s[HIDD * LATD];
  __shared__ float SWs[LATD * HIDD];
  const int b = blockIdx.x, t = threadIdx.x;
  for (int i = t; i < HIDD * LATD; i += 128) { W3s[i] = w3[i]; SWs[i] = skw[i]; }
  __syncthreads();
  const float* hrow = h2 + ((size_t)b * NPT + t) * HIDD;
  float lat[LATD];
#pragma unroll
  for (int c = 0; c < LATD; ++c) lat[c] = b3[c];
  for (int k = 0; k < HIDD; ++k) {
    const float hv = hrow[k];
#pragma unroll
    for (int c = 0; c < LATD; ++c) lat[c] = fmaf(hv, W3s[k * LATD + c], lat[c]);
  }
#pragma unroll
  for (int c = 0; c < LATD; ++c) latent[((size_t)b * NPT + t) * LATD + c] = lat[c];
  for (int c = 0; c < HIDD; ++c) {
    float acc = skb[c];
#pragma unroll
    for (int k = 0; k < LATD; ++k) acc = fmaf(lat[k], SWs[k * HIDD + c], acc);
    skip[((size_t)b * NPT + t) * HIDD + c] = acc;
  }
}

// ---------------------------------------------------------------------------
// Gabriel graph: adj (0/1 float) + edge_dist. grid = (B, 8), block = 256.
// Each block: 2048 (i,j) pairs, 128-point witness scan with points in LDS.
// ---------------------------------------------------------------------------
__global__ void gabriel_kernel(const float* __restrict__ latent,
                               float* __restrict__ adjf, float* __restrict__ edist) {
  __shared__ float P[NPT][4];
  const int b = blockIdx.x, t = threadIdx.x;
  for (int i = t; i < NPT * LATD; i += 256) P[i / 3][i % 3] = latent[(size_t)b * NPT * LATD + i];
  __syncthreads();
  const int base = blockIdx.y * 2048;
  for (int p = base + t; p < base + 2048; p += 256) {
    const int i = p >> 7, j = p & 127;
    const float mx = 0.5f * (P[i][0] + P[j][0]);
    const float my = 0.5f * (P[i][1] + P[j][1]);
    const float mz = 0.5f * (P[i][2] + P[j][2]);
    const float dx = P[i][0] - P[j][0];
    const float dy = P[i][1] - P[j][1];
    const float dz = P[i][2] - P[j][2];
    const float d2 = dx * dx + dy * dy + dz * dz;
    const float r2 = 0.25f * d2;
    bool blocked = false;
    for (int k = 0; k < NPT; ++k) {
      const float ex = P[k][0] - mx, ey = P[k][1] - my, ez = P[k][2] - mz;
      const float dd = ex * ex + ey * ey + ez * ez;
      blocked = blocked || ((dd < r2) && (k != i) && (k != j));
    }
    const bool adj = (!blocked) && (i != j);
    adjf[(size_t)b * NPT * NPT + p] = adj ? 1.0f : 0.0f;
    edist[(size_t)b * NPT * NPT + p] = adj ? sqrtf(d2) : 0.0f;
  }
}

// ---------------------------------------------------------------------------
// GAT layer 1 projections (Cin = 1): hl = x0*wl + bl ; hr = x0*wr + br
// x0 = latent[..., 2]. grid = B, block = 128
// ---------------------------------------------------------------------------
__global__ void gat1_proj_kernel(const float* __restrict__ latent,
                                 const float* __restrict__ wl, const float* __restrict__ bl,
                                 const float* __restrict__ wr, const float* __restrict__ br,
                                 float* __restrict__ hl, float* __restrict__ hr) {
  const int b = blockIdx.x, t = threadIdx.x;
  const float x0 = latent[((size_t)b * NPT + t) * LATD + 2];
  for (int c = 0; c < HIDD; ++c) {
    hl[((size_t)b * NPT + t) * HIDD + c] = fmaf(x0, wl[c], bl[c]);
    hr[((size_t)b * NPT + t) * HIDD + c] = fmaf(x0, wr[c], br[c]);
  }
}

// ---------------------------------------------------------------------------
// GATv2 attention: a[b,i,j] = softmax_j( mask( att . leaky(hr_i + hl_j) ) )
// grid = (B, 8), block = 128; each block handles 16 target rows i.
// ---------------------------------------------------------------------------
__global__ void gat_attn_kernel(const float* __restrict__ hl, const float* __restrict__ hr,
                                const float* __restrict__ att, const float* __restrict__ adjf,
                                float* __restrict__ a_out) {
  __shared__ float Hl[NPT * 65];
  __shared__ float attS[HIDD];
  __shared__ float hrRow[HIDD];
  __shared__ float red[NPT];
  const int b = blockIdx.x, t = threadIdx.x;
  for (int i = t; i < NPT * HIDD; i += 128)
    Hl[(i >> 6) * 65 + (i & 63)] = hl[(size_t)b * NPT * HIDD + i];
  if (t < HIDD) attS[t] = att[t];
  __syncthreads();

  const int i0 = blockIdx.y * 16;
  for (int ii = 0; ii < 16; ++ii) {
    const int i = i0 + ii;
    if (t < HIDD) hrRow[t] = hr[((size_t)b * NPT + i) * HIDD + t];
    __syncthreads();
    float s = 0.0f;
#pragma unroll 8
    for (int h = 0; h < HIDD; ++h) {
      float e = hrRow[h] + Hl[t * 65 + h];
      e = (e >= 0.0f) ? e : NEG_SL * e;
      s = fmaf(attS[h], e, s);
    }
    const bool valid = (adjf[(size_t)b * NPT * NPT + i * NPT + t] != 0.0f) || (i == t);
    const float sm = valid ? s : -1e30f;
    red[t] = sm; __syncthreads();
    for (int off = 64; off > 0; off >>= 1) {
      if (t < off) red[t] = fmaxf(red[t], red[t + off]);
      __syncthreads();
    }
    const float m = red[0]; __syncthreads();
    const float e = __expf(sm - m);
    red[t] = e; __syncthreads();
    for (int off = 64; off > 0; off >>= 1) {
      if (t < off) red[t] += red[t + off];
      __syncthreads();
    }
    const float denom = red[0]; __syncthreads();
    a_out[(size_t)b * NPT * NPT + i * NPT + t] = e / denom;
    __syncthreads();
  }
}

// ---------------------------------------------------------------------------
// Aggregation WMMA GEMM: out[b] = epi( a[b](128x128) @ hl[b](128x64) + bias
//                                      [+ alpha*skip] ), relu.  K = 128.
// grid = B, block = 256 (8 waves, one M-tile each). All 32 A fragments are
// preloaded from global into registers; k-loop fully unrolled so the LDS
// B-fragment loads batch ahead of the 4-accumulator WMMA chains.
// ---------------------------------------------------------------------------
template <bool SKIP>
__global__ __launch_bounds__(256) void gat_aggregate_wmma(
    const float* __restrict__ Amat, const float* __restrict__ Hlg,
    const float* __restrict__ bias, const float* __restrict__ skip,
    float alpha, float* __restrict__ out) {
  __shared__ float Hs[NPT * 65];
  __shared__ float Bs[HIDD];
  const int b = blockIdx.x, t = threadIdx.x;
  for (int i = t; i < NPT * HIDD; i += 256)
    Hs[(i >> 6) * 65 + (i & 63)] = Hlg[(size_t)b * NPT * HIDD + i];
  if (t < HIDD) Bs[t] = bias[t];

  const int w = t >> 5, lane = t & 31;
  const int rlo = lane & 15, half = lane >> 4;
  const float* Ab = Amat + (size_t)b * NPT * NPT;
  const int mt = w;                        // one 16-row M-tile per wave
  const int arow = mt * 16 + rlo;

  // Preload all 32 A fragments from global (overlaps with LDS fill above).
  v2f af[32];
#pragma unroll
  for (int kt = 0; kt < 32; ++kt) {
    const int kb = kt * 4 + half * 2;
    af[kt][0] = Ab[arow * NPT + kb + 0];
    af[kt][1] = Ab[arow * NPT + kb + 1];
  }
  __syncthreads();

  v8f c0, c1, c2, c3;
#pragma unroll
  for (int r = 0; r < 8; ++r) {
    c0[r] = Bs[ 0 + rlo];
    c1[r] = Bs[16 + rlo];
    c2[r] = Bs[32 + rlo];
    c3[r] = Bs[48 + rlo];
  }
#pragma unroll
  for (int kt = 0; kt < 32; ++kt) {
    const int kb = kt * 4 + half * 2;
    const int r0 = (kb + 0) * 65, r1 = (kb + 1) * 65;
    v2f b0, b1, b2, b3;
    b0[0] = Hs[r0 +  0 + rlo]; b0[1] = Hs[r1 +  0 + rlo];
    b1[0] = Hs[r0 + 16 + rlo]; b1[1] = Hs[r1 + 16 + rlo];
    b2[0] = Hs[r0 + 32 + rlo]; b2[1] = Hs[r1 + 32 + rlo];
    b3[0] = Hs[r0 + 48 + rlo]; b3[1] = Hs[r1 + 48 + rlo];
    c0 = __builtin_amdgcn_wmma_f32_16x16x4_f32(false, af[kt], false, b0, (short)0, c0, false, false);
    c1 = __builtin_amdgcn_wmma_f32_16x16x4_f32(false, af[kt], false, b1, (short)0, c1, false, false);
    c2 = __builtin_amdgcn_wmma_f32_16x16x4_f32(false, af[kt], false, b2, (short)0, c2, false, false);
    c3 = __builtin_amdgcn_wmma_f32_16x16x4_f32(false, af[kt], false, b3, (short)0, c3, false, false);
  }
#pragma unroll
  for (int r = 0; r < 8; ++r) {
    const int row = mt * 16 + half * 8 + r;
#pragma unroll
    for (int nt = 0; nt < 4; ++nt) {
      const int col = nt * 16 + rlo;
      float v = (nt == 0) ? c0[r] : (nt == 1) ? c1[r] : (nt == 2) ? c2[r] : c3[r];
      if (SKIP) v = fmaf(alpha, skip[((size_t)b * NPT + row) * HIDD + col], v);
      v = fmaxf(v, 0.0f);
      out[((size_t)b * NPT + row) * HIDD + col] = v;
    }
  }
}

// ---------------------------------------------------------------------------
// Output heads: out = x @ W(64 x ncol) + bias. grid = B, block = 128.
// ---------------------------------------------------------------------------
__global__ void head_kernel(const float* __restrict__ x, const float* __restrict__ W,
                            const float* __restrict__ bias, int ncol,
                            float* __restrict__ out) {
  const int b = blockIdx.x, t = threadIdx.x;
  const float* xr = x + ((size_t)b * NPT + t) * HIDD;
  for (int c = 0; c < ncol; ++c) {
    float acc = bias[c];
    for (int k = 0; k < HIDD; ++k) acc = fmaf(xr[k], W[k * ncol + c], acc);
    out[((size_t)b * NPT + t) * ncol + c] = acc;
  }
}

// ---------------------------------------------------------------------------
extern "C" void kernel_launch(void* const* d_in, const int* in_sizes, int n_in,
                              void* d_out, int out_size, void* d_ws, size_t ws_size,
                              hipStream_t stream) {
  const float* batch  = (const float*)d_in[0];
  const float* enc_w1 = (const float*)d_in[1];
  const float* enc_b1 = (const float*)d_in[2];
  const float* enc_w2 = (const float*)d_in[3];
  const float* enc_b2 = (const float*)d_in[4];
  const float* enc_w3 = (const float*)d_in[5];
  const float* enc_b3 = (const float*)d_in[6];
  const float* g_wl[4]  = {(const float*)d_in[7],  (const float*)d_in[13], (const float*)d_in[19], (const float*)d_in[25]};
  const float* g_bl[4]  = {(const float*)d_in[8],  (const float*)d_in[14], (const float*)d_in[20], (const float*)d_in[26]};
  const float* g_wr[4]  = {(const float*)d_in[9],  (const float*)d_in[15], (const float*)d_in[21], (const float*)d_in[27]};
  const float* g_br[4]  = {(const float*)d_in[10], (const float*)d_in[16], (const float*)d_in[22], (const float*)d_in[28]};
  const float* g_att[4] = {(const float*)d_in[11], (const float*)d_in[17], (const float*)d_in[23], (const float*)d_in[29]};
  const float* g_b[4]   = {(const float*)d_in[12], (const float*)d_in[18], (const float*)d_in[24], (const float*)d_in[30]};
  const float* lab_w  = (const float*)d_in[31];
  const float* lab_b  = (const float*)d_in[32];
  const float* val_w  = (const float*)d_in[33];
  const float* val_b  = (const float*)d_in[34];
  const float* skip_w = (const float*)d_in[35];
  const float* skip_b = (const float*)d_in[36];

  // d_out layout (floats, return order)
  float* out = (float*)d_out;
  float* o_feat4 = out;                 // B*N*4   = 32768
  float* o_feat1 = out + 32768;         // B*N*1   = 8192
  float* o_logit = out + 40960;         // B*N*4   = 32768
  float* o_value = out + 73728;         // B*N*1   = 8192
  float* o_lat   = out + 81920;         // B*N*3   = 24576
  float* o_adj   = out + 106496;        // B*N*N   = 1048576
  float* o_edist = out + 1155072;       // B*N*N   = 1048576

  // workspace layout (floats)
  float* ws = (float*)d_ws;
  const size_t NH = (size_t)BATCH * NPT * HIDD;   // 524288
  const size_t NA = (size_t)BATCH * NPT * NPT;    // 1048576
  float* ws_hl   = ws;
  float* ws_hr   = ws_hl + NH;
  float* ws_a    = ws_hr + NH;
  float* ws_skip = ws_a + NA;
  float* ws_x1   = ws_skip + NH;
  float* ws_x2   = ws_x1 + NH;
  float* ws_x34  = ws_x2 + NH;
  float* ws_h    = ws_x34 + NH;

  const dim3 gB(BATCH);
  const dim3 gB8(BATCH, 8);

  // Encoder
  enc1_kernel<<<gB, 128, 0, stream>>>(batch, enc_w1, enc_b1, ws_h, o_feat4, o_feat1);
  gemm_k64_wmma<true, false><<<gB, 256, 0, stream>>>(ws_h, enc_w2, enc_b2, nullptr, 0.0f, ws_x1); // h2
  latent_skip_kernel<<<gB, 128, 0, stream>>>(ws_x1, enc_w3, enc_b3, skip_w, skip_b, o_lat, ws_skip);

  // Gabriel graph (adj + edge_dist)
  gabriel_kernel<<<gB8, 256, 0, stream>>>(o_lat, o_adj, o_edist);

  // GAT layer 1 (Cin = 1)
  gat1_proj_kernel<<<gB, 128, 0, stream>>>(o_lat, g_wl[0], g_bl[0], g_wr[0], g_br[0], ws_hl, ws_hr);
  gat_attn_kernel<<<gB8, 128, 0, stream>>>(ws_hl, ws_hr, g_att[0], o_adj, ws_a);
  gat_aggregate_wmma<false><<<gB, 256, 0, stream>>>(ws_a, ws_hl, g_b[0], nullptr, 0.0f, ws_x1);

  // GAT layer 2
  gemm_k64_wmma<false, false><<<gB, 256, 0, stream>>>(ws_x1, g_wl[1], g_bl[1], nullptr, 0.0f, ws_hl);
  gemm_k64_wmma<false, false><<<gB, 256, 0, stream>>>(ws_x1, g_wr[1], g_br[1], nullptr, 0.0f, ws_hr);
  gat_attn_kernel<<<gB8, 128, 0, stream>>>(ws_hl, ws_hr, g_att[1], o_adj, ws_a);
  gat_aggregate_wmma<false><<<gB, 256, 0, stream>>>(ws_a, ws_hl, g_b[1], nullptr, 0.0f, ws_x2);

  // GAT layer 3 -> logits
  gemm_k64_wmma<false, false><<<gB, 256, 0, stream>>>(ws_x2, g_wl[2], g_bl[2], nullptr, 0.0f, ws_hl);
  gemm_k64_wmma<false, false><<<gB, 256, 0, stream>>>(ws_x2, g_wr[2], g_br[2], nullptr, 0.0f, ws_hr);
  gat_attn_kernel<<<gB8, 128, 0, stream>>>(ws_hl, ws_hr, g_att[2], o_adj, ws_a);
  gat_aggregate_wmma<true><<<gB, 256, 0, stream>>>(ws_a, ws_hl, g_b[2], ws_skip, ALPHA_C, ws_x34);
  head_kernel<<<gB, 128, 0, stream>>>(ws_x34, lab_w, lab_b, 4, o_logit);

  // GAT layer 4 -> values
  gemm_k64_wmma<false, false><<<gB, 256, 0, stream>>>(ws_x2, g_wl[3], g_bl[3], nullptr, 0.0f, ws_hl);
  gemm_k64_wmma<false, false><<<gB, 256, 0, stream>>>(ws_x2, g_wr[3], g_br[3], nullptr, 0.0f, ws_hr);
  gat_attn_kernel<<<gB8, 128, 0, stream>>>(ws_hl, ws_hr, g_att[3], o_adj, ws_a);
  gat_aggregate_wmma<true><<<gB, 256, 0, stream>>>(ws_a, ws_hl, g_b[3], ws_skip, ALPHA_C, ws_x34);
  head_kernel<<<gB, 128, 0, stream>>>(ws_x34, val_w, val_b, 1, o_value);
}